// InversePerspectiveProjection_72902774882475
// MI455X (gfx1250) — compile-verified
//
#include <hip/hip_runtime.h>

typedef __attribute__((ext_vector_type(2))) float v2f;
typedef __attribute__((ext_vector_type(8))) float v8f;

#define N_IMG   6
#define C_CH    64
#define H_IMG   224
#define W_IMG   400
#define HW_IMG  (H_IMG * W_IMG)        // 89600
#define CHW_IMG (C_CH * HW_IMG)        // 5734400
#define D_RES   8
#define H_RES   128
#define W_RES   128
#define P_TOT   (D_RES * H_RES * W_RES) // 131072
#define LDS_STRIDE 20                   // 16 cols (cams 0-4) + 3 (cam 5) + pad

__global__ __launch_bounds__(256)
void ipp_kernel(const float* __restrict__ images,
                const float* __restrict__ intr,   // (6,3,3)
                const float* __restrict__ rot,    // (6,3,3)
                const float* __restrict__ trans,  // (6,3)
                float* __restrict__ out)          // (64, 8,128,128)
{
    __shared__ float proj[256 * LDS_STRIDE];

    const int tid  = threadIdx.x;
    const int lane = tid & 31;
    const int wave = tid >> 5;
    const int Ncol = lane & 15;
    const bool hi  = lane >= 16;
    const int blockBase = blockIdx.x * 256;
    const int waveBase  = blockBase + wave * 32;

    // ---------------------------------------------------------------
    // Stage 1: projection of 32 voxels per wave via V_WMMA_F32_16X16X4_F32.
    // Combined transform: pix = M*pt + b, M = K@R (3x3), b = K@t.
    // B (4x16): column j = 3n+i holds [M[n][i][0..2], b[n][i]] for cams 0..4;
    // a second WMMA handles cam 5 in columns 0..2.
    // B VGPR layout (mirrors A 16x4): lanes 0-15 hold K=0,1; lanes 16-31 K=2,3.
    // ---------------------------------------------------------------
    v2f b1; b1.x = 0.0f; b1.y = 0.0f;
    if (Ncol < 15) {
        const int n = Ncol / 3;
        const int i = Ncol - 3 * n;
        const float* Kr = intr + n * 9 + i * 3;   // K[n][i][:]
        const float* Rn = rot  + n * 9;           // R[n][:][:]
        if (!hi) {
            b1.x = Kr[0]*Rn[0] + Kr[1]*Rn[3] + Kr[2]*Rn[6];   // M[i][0]
            b1.y = Kr[0]*Rn[1] + Kr[1]*Rn[4] + Kr[2]*Rn[7];   // M[i][1]
        } else {
            const float* Tn = trans + n * 3;
            b1.x = Kr[0]*Rn[2] + Kr[1]*Rn[5] + Kr[2]*Rn[8];   // M[i][2]
            b1.y = Kr[0]*Tn[0] + Kr[1]*Tn[1] + Kr[2]*Tn[2];   // b[i]
        }
    }
    v2f b2; b2.x = 0.0f; b2.y = 0.0f;
    if (Ncol < 3) {
        const int n = 5;
        const int i = Ncol;
        const float* Kr = intr + n * 9 + i * 3;
        const float* Rn = rot  + n * 9;
        if (!hi) {
            b2.x = Kr[0]*Rn[0] + Kr[1]*Rn[3] + Kr[2]*Rn[6];
            b2.y = Kr[0]*Rn[1] + Kr[1]*Rn[4] + Kr[2]*Rn[7];
        } else {
            const float* Tn = trans + n * 3;
            b2.x = Kr[0]*Rn[2] + Kr[1]*Rn[5] + Kr[2]*Rn[8];
            b2.y = Kr[0]*Tn[0] + Kr[1]*Tn[1] + Kr[2]*Tn[2];
        }
    }

    v8f cz = {0.f, 0.f, 0.f, 0.f, 0.f, 0.f, 0.f, 0.f};

#pragma unroll
    for (int t = 0; t < 2; ++t) {
        // A matrix 16x4 f32: lanes 0-15 carry {x,y} (K=0,1), lanes 16-31 {z,1} (K=2,3)
        const int pm  = waveBase + t * 16 + Ncol;   // point fed by this lane
        const int ixm = pm >> 10;                   // p = ix*(W*D) + iy*D + iz
        const int iym = (pm >> 3) & 127;
        const int izm = pm & 7;
        v2f a;
        if (!hi) { a.x = ((float)ixm - 63.5f) * 0.4f; a.y = ((float)iym - 63.5f) * 0.4f; }
        else     { a.x = ((float)izm - 3.5f)  * 0.4f; a.y = 1.0f; }

        v8f d1 = __builtin_amdgcn_wmma_f32_16x16x4_f32(false, a, false, b1,
                                                       (short)0, cz, false, false);
        v8f d2 = __builtin_amdgcn_wmma_f32_16x16x4_f32(false, a, false, b2,
                                                       (short)0, cz, false, false);

        // D layout: VGPR r, lanes 0-15 -> M=r, lanes 16-31 -> M=r+8; N = lane&15
        const int rowBase = wave * 32 + t * 16 + (hi ? 8 : 0);
#pragma unroll
        for (int r = 0; r < 8; ++r)
            proj[(rowBase + r) * LDS_STRIDE + Ncol] = d1[r];
        if (Ncol < 3) {
#pragma unroll
            for (int r = 0; r < 8; ++r)
                proj[(rowBase + r) * LDS_STRIDE + 16 + Ncol] = d2[r];
        }
    }
    __syncthreads();

    // Each lane reads its own point's LDS cols 0-17 (cams 0-4 {u,v,z}, zero col 15, cam 5 {u,v})
    float pj[18];
#pragma unroll
    for (int k = 0; k < 18; ++k)
        pj[k] = proj[tid * LDS_STRIDE + k];

    // ---------------------------------------------------------------
    // Stage 2: per-camera corner offsets + weights (validity and 1/6 folded in)
    // Reference quirk: xs derives from v = pix_y/z, ys from u = pix_x/z.
    // ---------------------------------------------------------------
    float wgt[6][4];
    int   off[6][4];
#pragma unroll
    for (int n = 0; n < 6; ++n) {
        const float u = pj[3*n + 0];
        const float v = pj[3*n + 1];
        const float z = pj[3*n + 2];
        const bool valid = (z >= 0.1f);
        const float uu = valid ? (u / z) : 0.0f;
        const float vv = valid ? (v / z) : 0.0f;
        // replicate reference op order exactly
        const float g0 = 2.0f * vv / 223.0f - 1.0f;
        const float g1 = 2.0f * uu / 399.0f - 1.0f;
        float xs = ((g0 + 1.0f) * 400.0f - 1.0f) * 0.5f;
        float ys = ((g1 + 1.0f) * 224.0f - 1.0f) * 0.5f;
        xs = fminf(fmaxf(xs, -10.0f), 410.0f);
        ys = fminf(fmaxf(ys, -10.0f), 234.0f);
        const float x0f = floorf(xs), y0f = floorf(ys);
        const float wx = xs - x0f,   wy = ys - y0f;
        const int x0 = (int)x0f, y0 = (int)y0f;
        const int x1 = x0 + 1,   y1 = y0 + 1;
        const int cx0 = min(max(x0, 0), W_IMG - 1);
        const int cx1 = min(max(x1, 0), W_IMG - 1);
        const int cy0 = min(max(y0, 0), H_IMG - 1);
        const int cy1 = min(max(y1, 0), H_IMG - 1);
        const bool x0in = (x0 >= 0) & (x0 < W_IMG);
        const bool x1in = (x1 >= 0) & (x1 < W_IMG);
        const bool y0in = (y0 >= 0) & (y0 < H_IMG);
        const bool y1in = (y1 >= 0) & (y1 < H_IMG);
        const float vf = valid ? (1.0f / 6.0f) : 0.0f;   // fold camera-mean
        const int nb = n * CHW_IMG;
        off[n][0] = nb + cy0 * W_IMG + cx0;
        off[n][1] = nb + cy0 * W_IMG + cx1;
        off[n][2] = nb + cy1 * W_IMG + cx0;
        off[n][3] = nb + cy1 * W_IMG + cx1;
        wgt[n][0] = (x0in && y0in) ? vf * (1.0f - wx) * (1.0f - wy) : 0.0f;
        wgt[n][1] = (x1in && y0in) ? vf * wx * (1.0f - wy)          : 0.0f;
        wgt[n][2] = (x0in && y1in) ? vf * (1.0f - wx) * wy          : 0.0f;
        wgt[n][3] = (x1in && y1in) ? vf * wx * wy                   : 0.0f;
    }

    // ---------------------------------------------------------------
    // Stage 3: channel loop — 24 unconditional clamped gathers + FMAs per c.
    // Images stay L2-resident; output uses NT stores to avoid evicting them.
    // ---------------------------------------------------------------
    const int p = blockBase + tid;
#pragma unroll 2
    for (int c = 0; c < C_CH; ++c) {
        const float* img_c = images + c * HW_IMG;
        float s = 0.0f;
#pragma unroll
        for (int n = 0; n < 6; ++n) {
            s += wgt[n][0] * img_c[off[n][0]];
            s += wgt[n][1] * img_c[off[n][1]];
            s += wgt[n][2] * img_c[off[n][2]];
            s += wgt[n][3] * img_c[off[n][3]];
        }
        __builtin_nontemporal_store(s, out + (size_t)c * P_TOT + p);
    }
}

extern "C" void kernel_launch(void* const* d_in, const int* in_sizes, int n_in,
                              void* d_out, int out_size, void* d_ws, size_t ws_size,
                              hipStream_t stream)
{
    const float* images = (const float*)d_in[0];
    const float* intr   = (const float*)d_in[1];
    const float* rot    = (const float*)d_in[2];
    const float* trans  = (const float*)d_in[3];
    float* out = (float*)d_out;

    dim3 grid(P_TOT / 256);   // 512 blocks x 256 threads = one thread per voxel
    dim3 block(256);
    ipp_kernel<<<grid, block, 0, stream>>>(images, intr, rot, trans, out);
}